// SPDTangentSpace_59227599012346
// MI455X (gfx1250) — compile-verified
//
#include <hip/hip_runtime.h>
#include <math.h>

// Batched SPD matrix log via inverse scaling-and-squaring + Newton-Schulz sqrt,
// all GEMMs on V_WMMA_F32_16X16X4_F32 (f32 matrix core), one 64x64 matrix per
// 128-thread (4-wave) workgroup, matrices resident in LDS.
//
// Exploits that every GEMM operand here (Y, Z, T, X, Q) is a polynomial of the
// same SPD matrix => symmetric. B fragments are therefore read transposed
// (B[k][n] == B[n][k]) so BOTH operand fragments are contiguous 8-byte LDS
// loads into adjacent VGPR pairs -> no register repacking before v_wmma.

typedef __attribute__((ext_vector_type(2))) float v2f;
typedef __attribute__((ext_vector_type(8))) float v8f;

#define MAT_N     64
#define STRIDE    68                  // pad: bank = (4*row + k) % 64 -> conflict-free
#define MAT_ELEMS (MAT_N * STRIDE)    // 4352 floats
#define NSQRT     5                   // number of matrix square roots (2^5 = 32)
#define NSITER    10                  // Newton-Schulz iterations per sqrt
#define QDEG      8                   // log(I+X) = X * sum_{i=0..QDEG} (-1)^i X^i/(i+1)

// One wave computes rows [16*wid, 16*wid+16) of D = Abuf * Bbuf (64x64x64, f32).
// Bbuf MUST be symmetric (true for all operands in this algorithm).
__device__ __forceinline__ void gemm_strip(const float* __restrict__ Abuf,
                                           const float* __restrict__ Bbuf,
                                           int wid, int lane, v8f acc[4]) {
  const int half = lane >> 4;        // lane half selects K=+0/+2 pair
  const int l16  = lane & 15;
  const int m    = (wid << 4) + l16; // A row held by this lane

  #pragma unroll
  for (int t = 0; t < 4; ++t) {
    #pragma unroll
    for (int g = 0; g < 8; ++g) acc[t][g] = 0.0f;
  }

  #pragma unroll 4
  for (int k0 = 0; k0 < MAT_N; k0 += 4) {
    const int ka = k0 + 2 * half;
    // A fragment 16x4: lane holds A[m][ka], A[m][ka+1] -> one ds_load_b64
    v2f afrag = *(const v2f*)(Abuf + m * STRIDE + ka);
    #pragma unroll
    for (int t = 0; t < 4; ++t) {
      const int n = (t << 4) + l16;
      // B fragment 4x16: needs B[ka][n], B[ka+1][n]; by symmetry read the row:
      // B[n][ka], B[n][ka+1] -> one ds_load_b64, already adjacent VGPR pair.
      v2f bfrag = *(const v2f*)(Bbuf + n * STRIDE + ka);
      acc[t] = __builtin_amdgcn_wmma_f32_16x16x4_f32(
          /*neg_a=*/false, afrag, /*neg_b=*/false, bfrag,
          /*c_mod=*/(short)0, acc[t], /*reuse_a=*/false, /*reuse_b=*/false);
    }
  }
}

// Store wave's strip to LDS:  D[m][n] = alpha*acc + diag*(m==n)
__device__ __forceinline__ void store_strip_lds(float* __restrict__ Dbuf, int wid, int lane,
                                                const v8f acc[4], float alpha, float diag) {
  const int half = lane >> 4, l16 = lane & 15;
  #pragma unroll
  for (int t = 0; t < 4; ++t) {
    const int n = (t << 4) + l16;
    #pragma unroll
    for (int g = 0; g < 8; ++g) {
      const int m = (wid << 4) + g + (half << 3);   // C/D layout: vgpr g -> M=g / g+8
      float v = alpha * acc[t][g];
      if (m == n) v += diag;
      Dbuf[m * STRIDE + n] = v;
    }
  }
}

__global__ __launch_bounds__(128)
void spd_logm_kernel(const float* __restrict__ A, float* __restrict__ out, int batch) {
  __shared__ float lds[4 * MAT_ELEMS];
  __shared__ float red;

  float* Y = lds;                    // current iterate (in-place safe: strip rows)
  float* T = lds + MAT_ELEMS;        // 3I - Z*Y   (later: X = B5 - I)
  // Z ping-pong buffers live at lds + (2+p)*MAT_ELEMS; computed per use (a
  // pointer ARRAY into LDS becomes an unlinkable constant initializer).

  const int b = blockIdx.x;
  if (b >= batch) return;
  const int tid  = threadIdx.x;
  const int lane = tid & 31;
  const int wid  = tid >> 5;

  // ---- Load A into LDS (padded) + Frobenius-norm reduction -----------------
  if (tid == 0) red = 0.0f;
  __syncthreads();
  const float4* in4 = (const float4*)(A + (size_t)b * (MAT_N * MAT_N));
  float s = 0.0f;
  #pragma unroll
  for (int q = 0; q < 8; ++q) {
    const int f   = tid + 128 * q;       // float4 index: coalesced
    const float4 v = in4[f];
    const int row = f >> 4;
    const int col = (f & 15) << 2;
    float* p = Y + row * STRIDE + col;
    p[0] = v.x; p[1] = v.y; p[2] = v.z; p[3] = v.w;
    s += v.x * v.x + v.y * v.y + v.z * v.z + v.w * v.w;
  }
  #pragma unroll
  for (int off = 16; off > 0; off >>= 1) s += __shfl_down(s, off, 32);
  if (lane == 0) atomicAdd(&red, s);
  __syncthreads();

  const float c0     = sqrtf(red);       // lambda_max(A) <= ||A||_F  -> spec(B) in (0,1]
  const float inv_c0 = 1.0f / c0;

  // Y = A / c0
  for (int e = tid; e < MAT_N * MAT_N; e += 128) {
    const int i = e >> 6, j = e & 63;
    Y[i * STRIDE + j] *= inv_c0;
  }

  int zp = 0;
  v8f acc[4];

  // ---- NSQRT levels of coupled Newton-Schulz square root -------------------
  for (int lvl = 0; lvl < NSQRT; ++lvl) {
    float* Zc = lds + (2 + zp) * MAT_ELEMS;
    // Z = I
    for (int e = tid; e < MAT_N * MAT_N; e += 128) {
      const int i = e >> 6, j = e & 63;
      Zc[i * STRIDE + j] = (i == j) ? 1.0f : 0.0f;
    }
    __syncthreads();

    for (int it = 0; it < NSITER; ++it) {
      float* Zcur = lds + (2 + zp) * MAT_ELEMS;
      float* Znew = lds + (2 + (zp ^ 1)) * MAT_ELEMS;
      // T = 3I - Z*Y
      gemm_strip(Zcur, Y, wid, lane, acc);
      store_strip_lds(T, wid, lane, acc, -1.0f, 3.0f);
      __syncthreads();
      // Y <- 0.5 * Y * T   (in place: wave reads only its own Y rows)
      gemm_strip(Y, T, wid, lane, acc);
      store_strip_lds(Y, wid, lane, acc, 0.5f, 0.0f);
      // Z' <- 0.5 * T * Z  (reads all Z rows -> ping-pong)
      gemm_strip(T, Zcur, wid, lane, acc);
      store_strip_lds(Znew, wid, lane, acc, 0.5f, 0.0f);
      __syncthreads();
      zp ^= 1;
    }
  }

  // ---- log phase: X = Y - I (spec(X) in [-0.11, 0]),  Q = Horner(Taylor) ---
  const float cTop = (((QDEG & 1) ? -1.0f : 1.0f) / (float)(QDEG + 1));
  {
    float* Qc = lds + (2 + zp) * MAT_ELEMS;
    for (int e = tid; e < MAT_N * MAT_N; e += 128) {
      const int i = e >> 6, j = e & 63;
      const float d = (i == j) ? 1.0f : 0.0f;
      T[i * STRIDE + j]  = Y[i * STRIDE + j] - d;  // X
      Qc[i * STRIDE + j] = d * cTop;               // Q = c_QDEG * I
    }
  }
  __syncthreads();

  for (int i = QDEG - 1; i >= 0; --i) {
    const float ci = ((i & 1) ? -1.0f : 1.0f) / (float)(i + 1);
    float* Qcur = lds + (2 + zp) * MAT_ELEMS;
    float* Qnew = lds + (2 + (zp ^ 1)) * MAT_ELEMS;
    gemm_strip(T, Qcur, wid, lane, acc);           // Q' = X*Q + c_i*I
    store_strip_lds(Qnew, wid, lane, acc, 1.0f, ci);
    __syncthreads();
    zp ^= 1;
  }

  // ---- out = 2^NSQRT * (X*Q) + log(c0)*I ----------------------------------
  gemm_strip(T, lds + (2 + zp) * MAT_ELEMS, wid, lane, acc);
  const float scale = (float)(1 << NSQRT);
  const float lc0   = logf(c0);
  float* o = out + (size_t)b * (MAT_N * MAT_N);
  {
    const int half = lane >> 4, l16 = lane & 15;
    #pragma unroll
    for (int t = 0; t < 4; ++t) {
      const int n = (t << 4) + l16;
      #pragma unroll
      for (int g = 0; g < 8; ++g) {
        const int m = (wid << 4) + g + (half << 3);
        float v = scale * acc[t][g];
        if (m == n) v += lc0;
        o[m * MAT_N + n] = v;
      }
    }
  }
}

extern "C" void kernel_launch(void* const* d_in, const int* in_sizes, int n_in,
                              void* d_out, int out_size, void* d_ws, size_t ws_size,
                              hipStream_t stream) {
  (void)n_in; (void)d_ws; (void)ws_size; (void)out_size;
  const float* A = (const float*)d_in[0];
  float* out     = (float*)d_out;
  const int batch = in_sizes[0] / (MAT_N * MAT_N);
  spd_logm_kernel<<<dim3(batch), dim3(128), 0, stream>>>(A, out, batch);
}